// TransportModule_79594333930098
// MI455X (gfx1250) — compile-verified
//
#include <hip/hip_runtime.h>
#include <hip/hip_bf16.h>

// Problem constants (fixed by the reference)
#define BQ    8
#define NPTS  16384
#define DDIM  64
#define PREAL 100
#define PPAD  112          // 7 tiles of 16
#define SORT_N 16384

typedef __attribute__((ext_vector_type(2))) float v2f;
typedef __attribute__((ext_vector_type(8))) float v8f;

// ---------------------------------------------------------------------------
// Kernel 1: normalize theta rows -> theta_n [PPAD, DDIM], pad rows = 0
// One wave (32 lanes) per row; each lane owns 2 of the 64 elements.
// ---------------------------------------------------------------------------
__global__ void norm_theta_kernel(const float* __restrict__ theta,
                                  float* __restrict__ thn) {
    int p    = blockIdx.x;        // 0..PPAD-1
    int lane = threadIdx.x;       // 0..31
    float v0 = 0.f, v1 = 0.f;
    if (p < PREAL) {
        v0 = theta[p * DDIM + lane];
        v1 = theta[p * DDIM + lane + 32];
    }
    float s = v0 * v0 + v1 * v1;
    #pragma unroll
    for (int off = 16; off > 0; off >>= 1)
        s += __shfl_xor(s, off, 32);
    float r = (p < PREAL) ? rsqrtf(s) : 0.0f;
    thn[p * DDIM + lane]      = v0 * r;
    thn[p * DDIM + lane + 32] = v1 * r;
}

// ---------------------------------------------------------------------------
// Kernel 2: projections. For each batch b:
//   xproj[b,p,n] = sum_d thn[p,d] * x[b,n,d]   (same for y)
// One wave computes one 16(P) x 16(N) output tile for BOTH x and y,
// using V_WMMA_F32_16X16X4_F32 over K = DDIM = 64 (16 k-steps).
//
// f32 16x16x4 VGPR layout (wave32):
//   A (16x4, M x K): lanes 0-15 hold M=lane (K=+0/+1 in v0/v1),
//                    lanes 16-31 hold M=lane-16 (K=+2/+3 in v0/v1)
//   B (4x16, K x N): mirrored with N across lanes.
//   C/D (16x16): VGPR j -> M=j (lanes 0-15) / M=j+8 (lanes 16-31), N=lane%16
// ---------------------------------------------------------------------------
__global__ void proj_kernel(const float* __restrict__ x,
                            const float* __restrict__ y,
                            const float* __restrict__ thn,
                            float* __restrict__ xproj,
                            float* __restrict__ yproj) {
    const int NT_N = NPTS / 16;                              // 1024 n-tiles
    int wave = (int)((blockIdx.x * blockDim.x + threadIdx.x) >> 5);
    int lane = threadIdx.x & 31;
    int nt   = wave % NT_N;
    int pt   = wave / NT_N;                                  // 0 .. BQ*7-1
    int b    = pt / (PPAD / 16);
    int p0   = (pt % (PPAD / 16)) * 16;
    int n0   = nt * 16;

    int col   = lane & 15;
    int khalf = (lane >> 4) << 1;                            // 0 or 2

    const float* arow  = thn + (size_t)(p0 + col) * DDIM + khalf;
    const float* bxrow = x + ((size_t)b * NPTS + n0 + col) * DDIM + khalf;
    const float* byrow = y + ((size_t)b * NPTS + n0 + col) * DDIM + khalf;

    v8f accx = {};
    v8f accy = {};
    #pragma unroll
    for (int k = 0; k < DDIM; k += 4) {
        v2f a  = *(const v2f*)(arow + k);
        v2f bx = *(const v2f*)(bxrow + k);
        v2f by = *(const v2f*)(byrow + k);
        accx = __builtin_amdgcn_wmma_f32_16x16x4_f32(false, a, false, bx,
                                                     (short)0, accx, false, false);
        accy = __builtin_amdgcn_wmma_f32_16x16x4_f32(false, a, false, by,
                                                     (short)0, accy, false, false);
    }

    int mhi = (lane >> 4) << 3;                              // +8 for upper half
    #pragma unroll
    for (int j = 0; j < 8; ++j) {
        size_t off = ((size_t)(b * PPAD + p0 + j + mhi)) * NPTS + n0 + col;
        xproj[off] = accx[j];
        yproj[off] = accy[j];
    }
}

// ---------------------------------------------------------------------------
// Kernel 3: per-(b,p) bitonic sorts in LDS + scatter of the OT diff.
//   phase 1: sort (x_proj row, index) ascending  -> idx[i] = argsort(x)[i]
//   phase 2: sort y_proj row ascending (keys only)
//   phase 3: diff[idx[i]] = y_sorted[i] - x_proj[idx[i]]  (in-place over xproj)
// Dynamic LDS: 16384 f32 keys (64KB) + 16384 u16 idx (32KB) = 96KB.
// ---------------------------------------------------------------------------
template <bool WITH_IDX>
__device__ inline void bitonic_sort(float* keys, unsigned short* idx,
                                    int tid, int nthreads) {
    for (int k = 2; k <= SORT_N; k <<= 1) {
        for (int j = k >> 1; j > 0; j >>= 1) {
            int mask = j - 1;
            for (int t = tid; t < SORT_N / 2; t += nthreads) {
                int i   = ((t & ~mask) << 1) | (t & mask);   // insert 0 at bit j
                int ixj = i | j;
                float a = keys[i];
                float c = keys[ixj];
                bool up = ((i & k) == 0);
                if ((a > c) == up) {
                    keys[i]   = c;
                    keys[ixj] = a;
                    if (WITH_IDX) {
                        unsigned short tI = idx[i];
                        idx[i]   = idx[ixj];
                        idx[ixj] = tI;
                    }
                }
            }
            __syncthreads();
        }
    }
}

__global__ void sort_row_kernel(float* __restrict__ xproj,          // becomes diff
                                const float* __restrict__ yproj) {
    extern __shared__ char smem[];
    float* keys         = (float*)smem;
    unsigned short* idx = (unsigned short*)(smem + (size_t)SORT_N * 4);

    int tid      = threadIdx.x;
    int nthreads = blockDim.x;
    int row      = blockIdx.x;                // 0 .. BQ*PREAL-1
    int b        = row / PREAL;
    int p        = row % PREAL;
    size_t base  = ((size_t)(b * PPAD + p)) * NPTS;

    // phase 1: sort x keys with index payload
    for (int i = tid; i < SORT_N; i += nthreads) {
        keys[i] = xproj[base + i];
        idx[i]  = (unsigned short)i;
    }
    __syncthreads();
    bitonic_sort<true>(keys, idx, tid, nthreads);

    // phase 2: sort y keys (payload untouched)
    for (int i = tid; i < SORT_N; i += nthreads)
        keys[i] = yproj[base + i];
    __syncthreads();
    bitonic_sort<false>(keys, idx, tid, nthreads);

    // phase 3: scatter diff = y_sorted[rank] - x_proj, in place
    for (int i = tid; i < SORT_N; i += nthreads) {
        int j    = idx[i];
        float yv = keys[i];
        float xv = xproj[base + j];
        xproj[base + j] = yv - xv;
    }
}

// ---------------------------------------------------------------------------
// Kernel 4: back-projection + residual add:
//   out[b,n,d] = x[b,n,d] + (1/PREAL) * sum_p diff[b,p,n] * thn[p,d]
// One wave per 16(N) x 16(D) tile; K = PPAD = 112 (28 k-steps of 4).
// Padded rows of diff and thn are exactly zero so they contribute nothing.
// ---------------------------------------------------------------------------
__global__ void backproj_kernel(const float* __restrict__ diff,
                                const float* __restrict__ thn,
                                const float* __restrict__ xin,
                                float* __restrict__ out) {
    const int NT_M = NPTS / 16;                              // 1024
    const int NT_D = DDIM / 16;                              // 4
    int wave = (int)((blockIdx.x * blockDim.x + threadIdx.x) >> 5);
    int lane = threadIdx.x & 31;

    int dt  = wave % NT_D;
    int rem = wave / NT_D;
    int mt  = rem % NT_M;
    int b   = rem / NT_M;
    int m0  = mt * 16;
    int d0  = dt * 16;

    int col   = lane & 15;
    int khalf = (lane >> 4) << 1;

    v8f acc = {};
    #pragma unroll 4
    for (int k = 0; k < PPAD; k += 4) {
        // A (16x4): A[m][kk] = diff[b, k+khalf+{0,1}, m0+m]
        size_t abase = ((size_t)(b * PPAD + k + khalf)) * NPTS + m0 + col;
        v2f a;
        a.x = diff[abase];
        a.y = diff[abase + NPTS];
        // B (4x16): B[kk][d] = thn[(k+khalf+{0,1})*DDIM + d0+d]
        const float* bb = thn + (size_t)(k + khalf) * DDIM + d0 + col;
        v2f bt;
        bt.x = bb[0];
        bt.y = bb[DDIM];
        acc = __builtin_amdgcn_wmma_f32_16x16x4_f32(false, a, false, bt,
                                                    (short)0, acc, false, false);
    }

    const float invP = 1.0f / (float)PREAL;
    int mhi = (lane >> 4) << 3;
    #pragma unroll
    for (int j = 0; j < 8; ++j) {
        size_t off = ((size_t)b * NPTS + m0 + j + mhi) * DDIM + d0 + col;
        out[off] = xin[off] + acc[j] * invP;
    }
}

// ---------------------------------------------------------------------------
// Launch: theta_n | xproj(->diff) | yproj live in d_ws.
// ---------------------------------------------------------------------------
extern "C" void kernel_launch(void* const* d_in, const int* in_sizes, int n_in,
                              void* d_out, int out_size, void* d_ws, size_t ws_size,
                              hipStream_t stream) {
    const float* x     = (const float*)d_in[0];
    const float* y     = (const float*)d_in[1];
    const float* theta = (const float*)d_in[2];
    float* out = (float*)d_out;

    char* ws = (char*)d_ws;
    const size_t THN_BYTES  = (size_t)PPAD * DDIM * sizeof(float);        // 28 KB
    const size_t PROJ_BYTES = (size_t)BQ * PPAD * NPTS * sizeof(float);   // ~58.7 MB
    float* thn   = (float*)ws;
    float* xproj = (float*)(ws + THN_BYTES);
    float* yproj = (float*)(ws + THN_BYTES + PROJ_BYTES);

    // 1) normalize projection directions (zero-padded to PPAD rows)
    norm_theta_kernel<<<PPAD, 32, 0, stream>>>(theta, thn);

    // 2) projection GEMMs: B * (PPAD/16) * (NPTS/16) waves, 8 waves/block
    {
        int waves  = BQ * (PPAD / 16) * (NPTS / 16);   // 57344
        int blocks = waves / 8;                         // 7168
        proj_kernel<<<blocks, 256, 0, stream>>>(x, y, thn, xproj, yproj);
    }

    // 3) per-(b,p) sorts + diff scatter: 800 blocks, 96KB dynamic LDS each
    {
        size_t lds = (size_t)SORT_N * sizeof(float) +
                     (size_t)SORT_N * sizeof(unsigned short);  // 98304 B
        sort_row_kernel<<<BQ * PREAL, 1024, lds, stream>>>(xproj, yproj);
    }

    // 4) back-projection + residual add: B * 1024 * 4 waves, 8 waves/block
    {
        int waves  = BQ * (NPTS / 16) * (DDIM / 16);   // 32768
        int blocks = waves / 8;                         // 4096
        backproj_kernel<<<blocks, 256, 0, stream>>>(xproj, thn, x, out);
    }
}